// MultiBoxLoss_29686813950148
// MI455X (gfx1250) — compile-verified
//
#include <hip/hip_runtime.h>
#include <math.h>

// MultiBoxLoss for MI455X (gfx1250, wave32).
// B=128, P=8732, C=21, N=32. Output: {loss_l/Npos, loss_c/Npos}.

typedef __attribute__((ext_vector_type(2))) float v2f;
typedef __attribute__((ext_vector_type(8))) float v8f;

#define B_SZ 128
#define P_SZ 8732
#define C_SZ 21
#define N_GT 32

// LDS byte-offset of a __shared__ object (generic -> AS(3) -> int)
#define LDS_OFF(p) ((unsigned)(unsigned long long)(__attribute__((address_space(3))) void*)(p))

// Async memory->LDS copy of one 16B chunk (GVS mode: SADDR base + 32b VGPR off)
__device__ __forceinline__ void async_g2lds_b128(unsigned ldsoff,
                                                 unsigned voff,
                                                 unsigned long long gbase) {
  asm volatile("global_load_async_to_lds_b128 %0, %1, %2"
               :: "v"(ldsoff), "v"(voff), "s"(gbase)
               : "memory");
}
__device__ __forceinline__ void wait_async0() {
  asm volatile("s_wait_asynccnt 0" ::: "memory");
}

// ---------------------------------------------------------------- zero init
__global__ void zero_ws_kernel(unsigned long long* key_bp, int* num_pos, float* acc) {
  int t = blockIdx.x * blockDim.x + threadIdx.x;
  int stride = gridDim.x * blockDim.x;
  for (int i = t; i < B_SZ * N_GT; i += stride) key_bp[i] = 0ull;
  if (t < B_SZ) num_pos[t] = 0;
  if (t < 2) acc[t] = 0.f;
}

// ---------------------------------------------------------------- matching
// One thread per (b, prior). Computes IoU vs all 32 GT boxes.
// best-truth (max over GTs, first-index ties) stored per prior.
// best-prior (max over priors, first-index ties) via u64 key atomicMax:
//   key = bits(iou)<<32 | (0xFFFFFFFF - p)   (iou>=0 so bit order == value order)
__global__ void match_kernel(const float* __restrict__ dbox,
                             const float* __restrict__ targets,
                             float* __restrict__ bt_ov, int* __restrict__ bt_idx,
                             unsigned long long* __restrict__ key_bp) {
  __shared__ float gt[N_GT * 5];
  const int b = blockIdx.y;
  const int p = blockIdx.x * 256 + threadIdx.x;
  if (threadIdx.x < N_GT * 5) gt[threadIdx.x] = targets[b * N_GT * 5 + threadIdx.x];
  __syncthreads();

  float px1 = 0.f, py1 = 0.f, px2 = 0.f, py2 = 0.f, parea = 1.f;
  const bool valid = (p < P_SZ);
  if (valid) {
    float cx = dbox[p * 4 + 0], cy = dbox[p * 4 + 1];
    float w  = dbox[p * 4 + 2], h  = dbox[p * 4 + 3];
    px1 = cx - 0.5f * w; py1 = cy - 0.5f * h;
    px2 = cx + 0.5f * w; py2 = cy + 0.5f * h;
    parea = w * h;
  }
  float best_ov = -1.f; int best_n = 0;
  for (int n = 0; n < N_GT; ++n) {
    float bx1 = gt[n * 5 + 0], by1 = gt[n * 5 + 1];
    float bx2 = gt[n * 5 + 2], by2 = gt[n * 5 + 3];
    float iw = fmaxf(fminf(px2, bx2) - fmaxf(px1, bx1), 0.f);
    float ih = fmaxf(fminf(py2, by2) - fmaxf(py1, by1), 0.f);
    float inter = iw * ih;
    float ba = (bx2 - bx1) * (by2 - by1);
    float iou = inter / (parea + ba - inter);
    if (valid && iou > best_ov) { best_ov = iou; best_n = n; }
    unsigned long long key = valid
        ? ((unsigned long long)__float_as_uint(iou) << 32) |
          (unsigned long long)(0xFFFFFFFFu - (unsigned)p)
        : 0ull;
    // wave32 max-reduction
    for (int off = 16; off > 0; off >>= 1) {
      unsigned long long o = __shfl_xor(key, off);
      key = (key > o) ? key : o;
    }
    if ((threadIdx.x & 31) == 0) atomicMax(&key_bp[b * N_GT + n], key);
  }
  if (valid) {
    bt_ov[(size_t)b * P_SZ + p] = best_ov;
    bt_idx[(size_t)b * P_SZ + p] = best_n;
  }
}

// Sequential per-batch override so duplicate best-prior indices get
// last-writer-wins semantics like the reference scatter.
__global__ void override_kernel(const unsigned long long* __restrict__ key_bp,
                                float* __restrict__ bt_ov, int* __restrict__ bt_idx) {
  const int b = blockIdx.x;
  if (threadIdx.x != 0) return;
  for (int n = 0; n < N_GT; ++n) {
    unsigned long long key = key_bp[b * N_GT + n];
    unsigned p = 0xFFFFFFFFu - (unsigned)(key & 0xFFFFFFFFull);
    bt_ov[(size_t)b * P_SZ + p] = 2.0f;
    bt_idx[(size_t)b * P_SZ + p] = n;
  }
}

// ---------------------------------------------------------------- main loss
// Block = 256 threads (8 waves), handles 128 priors of one batch row.
// conf tile DMA'd memory->LDS with GLOBAL_LOAD_ASYNC_TO_LDS_B128 (ASYNCcnt),
// overlapping the per-prior loc-loss phase. Base address and tile byte counts
// are provably 16B-aligned (rows%4==0 for both 128- and 28-row tiles).
// Each wave computes 16 rows' sum-of-exp with chained V_WMMA_F32_16X16X4_F32:
//   A (16x4 f32): lanes0-15 hold {K0,K1}, lanes16-31 hold {K2,K3} of rows 0-15
//   B = all ones  => every D column = row sum
//   D (16x16 f32): lanes0-15 elem k = rowsum(k); lanes16-31 elem k = rowsum(k+8)
__global__ __launch_bounds__(256) void loss_kernel(
    const float* __restrict__ loc_data, const float* __restrict__ conf_data,
    const float* __restrict__ dbox, const float* __restrict__ targets,
    const float* __restrict__ bt_ov, const int* __restrict__ bt_idx,
    float* __restrict__ ce_neg, int* __restrict__ num_pos,
    float* __restrict__ acc) {
  __shared__ __align__(16) float conf_s[128 * C_SZ];
  __shared__ int   conft_s[128];
  __shared__ int   pos_s[128];
  __shared__ float maxv_s[128];
  __shared__ float sums_s[8][16];
  __shared__ float red_l, red_c;
  __shared__ int   red_p;

  const int b = blockIdx.y;
  const int p0 = blockIdx.x * 128;
  const int tid = threadIdx.x;
  if (tid == 0) { red_l = 0.f; red_c = 0.f; red_p = 0; }

  // Phase A: async DMA of conf tile directly into LDS (no VGPR round-trip)
  {
    const int rows = min(128, P_SZ - p0);           // 128 or 28; rows%4 == 0
    const int nchunks = (rows * C_SZ) >> 2;         // 16B chunks
    const unsigned lds_base = LDS_OFF(conf_s);
    const unsigned long long gbase = (unsigned long long)conf_data +
        (unsigned long long)(b * P_SZ + p0) * (C_SZ * 4);
    for (int c = tid; c < nchunks; c += 256)
      async_g2lds_b128(lds_base + ((unsigned)c << 4), (unsigned)c << 4, gbase);
    // zero-fill the OOB tail rows (disjoint LDS range; plain stores)
    for (int i = rows * C_SZ + tid; i < 128 * C_SZ; i += 256) conf_s[i] = 0.f;
  }

  // Phase B overlaps the DMA: per-prior conf_t / pos / smooth-L1 loc loss
  if (tid < 128) {
    const int r = tid, p = p0 + r;
    int ct = 0, posf = 0;
    float sl1 = 0.f;
    if (p < P_SZ) {
      float ov = bt_ov[(size_t)b * P_SZ + p];
      int n = bt_idx[(size_t)b * P_SZ + p];
      const float* t5 = &targets[(size_t)(b * N_GT + n) * 5];
      ct = (ov < 0.5f) ? 0 : ((int)t5[4] + 1);
      posf = (ct > 0);
      if (posf) {
        float x1 = t5[0], y1 = t5[1], x2 = t5[2], y2 = t5[3];
        float mcx = 0.5f * (x1 + x2), mcy = 0.5f * (y1 + y2);
        float mw = x2 - x1, mh = y2 - y1;
        float pcx = dbox[p * 4 + 0], pcy = dbox[p * 4 + 1];
        float pw  = dbox[p * 4 + 2], ph  = dbox[p * 4 + 3];
        float g0 = (mcx - pcx) / (0.1f * pw);
        float g1 = (mcy - pcy) / (0.1f * ph);
        float g2 = __logf(mw / pw) * 5.0f;   // /0.2
        float g3 = __logf(mh / ph) * 5.0f;
        const float* ld = &loc_data[(size_t)(b * P_SZ + p) * 4];
        float d0 = ld[0] - g0, d1 = ld[1] - g1, d2 = ld[2] - g2, d3 = ld[3] - g3;
        #define SL1(d) (fabsf(d) < 1.f ? 0.5f * (d) * (d) : fabsf(d) - 0.5f)
        sl1 = SL1(d0) + SL1(d1) + SL1(d2) + SL1(d3);
        #undef SL1
      }
    }
    conft_s[r] = ct; pos_s[r] = posf;
    if (posf) { atomicAdd(&red_l, sl1); atomicAdd(&red_p, 1); }
  }
  wait_async0();      // this wave's DMA chunks have landed in LDS
  __syncthreads();    // all waves' chunks visible block-wide

  // Phase C: WMMA logsumexp denominators, 16 rows per wave (EXEC all ones)
  {
    const int wave = tid >> 5, lane = tid & 31;
    const int L = lane & 15, half = lane >> 4;
    const int r = wave * 16 + L;
    float m = -1e30f;
    const int cb = half ? 11 : 0, cend = half ? C_SZ : 11;
    for (int c = cb; c < cend; ++c) m = fmaxf(m, conf_s[r * C_SZ + c]);
    m = fmaxf(m, __shfl_xor(m, 16));   // both halves now hold row max
    if (!half) maxv_s[r] = m;

    v8f acc8 = {0.f, 0.f, 0.f, 0.f, 0.f, 0.f, 0.f, 0.f};
    v2f ones = {1.f, 1.f};
    #pragma unroll
    for (int chunk = 0; chunk < 6; ++chunk) {
      int c0 = chunk * 4 + half * 2;           // K-slot classes for this lane
      float e0 = (c0     < C_SZ) ? __expf(conf_s[r * C_SZ + c0]     - m) : 0.f;
      float e1 = (c0 + 1 < C_SZ) ? __expf(conf_s[r * C_SZ + c0 + 1] - m) : 0.f;
      v2f a = {e0, e1};
      acc8 = __builtin_amdgcn_wmma_f32_16x16x4_f32(
          false, a, false, ones, (short)0, acc8, false, false);
    }
    if (lane == 0) {
      #pragma unroll
      for (int k = 0; k < 8; ++k) sums_s[wave][k] = acc8[k];
    }
    if (lane == 16) {
      #pragma unroll
      for (int k = 0; k < 8; ++k) sums_s[wave][8 + k] = acc8[k];
    }
  }
  __syncthreads();

  // Phase D: CE per prior; positives accumulate, negatives go to ce_neg
  if (tid < 128) {
    const int r = tid, p = p0 + r;
    if (p < P_SZ) {
      float lse = maxv_s[r] + __logf(sums_s[r >> 4][r & 15]);
      int ct = conft_s[r];
      float cev = lse - conf_s[r * C_SZ + ct];
      int posf = pos_s[r];
      ce_neg[(size_t)b * P_SZ + p] = posf ? 0.f : cev;
      if (posf) atomicAdd(&red_c, cev);
    }
  }
  __syncthreads();
  if (tid == 0) {
    if (red_l != 0.f) atomicAdd(&acc[0], red_l);
    if (red_c != 0.f) atomicAdd(&acc[1], red_c);
    if (red_p != 0)  atomicAdd(&num_pos[b], red_p);
  }
}

// ---------------------------------------------------------------- top-k sum
// One block per batch row. Row (34928B, 16B-aligned) DMA'd into LDS with
// async B128 loads, then exact sum of k largest ce_neg values via 32-step
// binary search on the uint bit pattern (all values >= 0).
__global__ __launch_bounds__(1024) void topk_kernel(
    const float* __restrict__ ce_neg, const int* __restrict__ num_pos,
    float* __restrict__ acc) {
  __shared__ __align__(16) unsigned vals[P_SZ];
  __shared__ int cnt_s;
  __shared__ float fsum_s;
  __shared__ int cg_s;
  const int b = blockIdx.x;
  const int tid = threadIdx.x;
  {
    const unsigned lds_base = LDS_OFF(vals);
    const unsigned long long gbase = (unsigned long long)ce_neg +
        (unsigned long long)b * P_SZ * 4;
    const int nchunks = (P_SZ * 4) >> 4;   // 2183 exact
    for (int c = tid; c < nchunks; c += 1024)
      async_g2lds_b128(lds_base + ((unsigned)c << 4), (unsigned)c << 4, gbase);
  }
  const int np = num_pos[b];
  long long kk = 3LL * np;
  const int k = (kk > (P_SZ - 1)) ? (P_SZ - 1) : (int)kk;
  wait_async0();
  __syncthreads();
  if (k <= 0) return;  // uniform exit

  unsigned lo = 0u, hi = 0xFFFFFFFFu;
  while (lo < hi) {    // uniform loop
    unsigned d = hi - lo;
    unsigned mid = lo + (d >> 1) + (d & 1u);   // ceil midpoint, no overflow
    __syncthreads();
    if (tid == 0) cnt_s = 0;
    __syncthreads();
    int local = 0;
    for (int i = tid; i < P_SZ; i += 1024) local += (vals[i] >= mid);
    if (local) atomicAdd(&cnt_s, local);
    __syncthreads();
    int cnt = cnt_s;
    if (cnt >= k) lo = mid; else hi = mid - 1;
  }
  const unsigned t = lo;  // k-th largest value (present in array)
  __syncthreads();
  if (tid == 0) { fsum_s = 0.f; cg_s = 0; }
  __syncthreads();
  float ls = 0.f; int lc = 0;
  for (int i = tid; i < P_SZ; i += 1024) {
    unsigned v = vals[i];
    if (v > t) { ls += __uint_as_float(v); lc++; }
  }
  if (lc) { atomicAdd(&fsum_s, ls); atomicAdd(&cg_s, lc); }
  __syncthreads();
  if (tid == 0) {
    float topk = fsum_s + (float)(k - cg_s) * __uint_as_float(t);
    atomicAdd(&acc[1], topk);
  }
}

// ---------------------------------------------------------------- finalize
__global__ void finalize_kernel(const float* __restrict__ acc,
                                const int* __restrict__ num_pos,
                                float* __restrict__ out) {
  if (threadIdx.x == 0 && blockIdx.x == 0) {
    int tp = 0;
    for (int b = 0; b < B_SZ; ++b) tp += num_pos[b];
    float Nf = fmaxf((float)tp, 1.f);
    out[0] = acc[0] / Nf;
    out[1] = acc[1] / Nf;
  }
}

// ---------------------------------------------------------------- launch
extern "C" void kernel_launch(void* const* d_in, const int* in_sizes, int n_in,
                              void* d_out, int out_size, void* d_ws, size_t ws_size,
                              hipStream_t stream) {
  const float* loc     = (const float*)d_in[0];  // (128, 8732, 4)
  const float* conf    = (const float*)d_in[1];  // (128, 8732, 21)
  const float* dbox    = (const float*)d_in[2];  // (8732, 4)
  const float* targets = (const float*)d_in[3];  // (128, 32, 5)

  char* ws = (char*)d_ws;
  size_t off = 0;
  auto alloc = [&](size_t bytes) -> void* {
    void* p = ws + off;
    off = (off + bytes + 255) & ~(size_t)255;
    return p;
  };
  unsigned long long* key_bp = (unsigned long long*)alloc((size_t)B_SZ * N_GT * 8);
  float* bt_ov  = (float*)alloc((size_t)B_SZ * P_SZ * 4);
  int*   bt_idx = (int*)  alloc((size_t)B_SZ * P_SZ * 4);
  float* ce_neg = (float*)alloc((size_t)B_SZ * P_SZ * 4);
  int*   num_pos = (int*) alloc((size_t)B_SZ * 4);
  float* acc     = (float*)alloc(64);

  zero_ws_kernel<<<4, 1024, 0, stream>>>(key_bp, num_pos, acc);
  match_kernel<<<dim3((P_SZ + 255) / 256, B_SZ), 256, 0, stream>>>(
      dbox, targets, bt_ov, bt_idx, key_bp);
  override_kernel<<<B_SZ, 32, 0, stream>>>(key_bp, bt_ov, bt_idx);
  loss_kernel<<<dim3((P_SZ + 127) / 128, B_SZ), 256, 0, stream>>>(
      loc, conf, dbox, targets, bt_ov, bt_idx, ce_neg, num_pos, acc);
  topk_kernel<<<B_SZ, 1024, 0, stream>>>(ce_neg, num_pos, acc);
  finalize_kernel<<<1, 32, 0, stream>>>(acc, num_pos, (float*)d_out);
}